// DenseRNN_61942018342934
// MI455X (gfx1250) — compile-verified
//
#include <hip/hip_runtime.h>
#include <cmath>

// ---------------------------------------------------------------------------
// DenseRNN (Householder-reflection linear RNN) for MI455X / gfx1250
//   xz = x @ Win^T ; xi,z = split(xz)
//   lam = exp(-8*softplus(omega)*sigmoid(xi@Wl^T + bl))
//   v   = xi@Wv^T + bv
//   scan: h_t = lam_t*(h_{t-1} - 2 v_t (v.h)/(v.v)) + (1-lam_t)*xi_t ; h_0 = xi_0
//   out = (silu(z)*h) @ Wout^T
// GEMMs: V_WMMA_F32_16X16X4_F32 + double-buffered async global->LDS staging.
// ---------------------------------------------------------------------------

typedef __attribute__((ext_vector_type(2))) float v2f;
typedef __attribute__((ext_vector_type(8))) float v8f;
typedef int v4i_gcc __attribute__((vector_size(16)));   // matches builtin pointee

#define D_MODEL   1024
#define D_INNER   2048
#define BATCH     4
#define SEQ       2048
#define M_ROWS    (BATCH * SEQ)        // 8192

// GEMM tiling: 128 threads = 4 waves; each wave owns a 32x64 C strip.
#define TM 128        // block rows
#define TN 64         // block cols
#define TK 32         // K tile
#define LDS_PAD 4     // row stride 36 floats = 144 B (16B aligned for b128 async)
#define LDS_LD  (TK + LDS_PAD)

// Epilogue modes
#define EPI_PLAIN  0   // out0[m,col] = acc (+ bias if non-null)
#define EPI_LAMBDA 1   // out0[m,col] = exp(-8*softplus(omega[col])*sigmoid(acc+bias[col]))
#define EPI_SPLIT  2   // col < D_INNER -> out0 (xi), else -> out1 (z), both ld = D_INNER

// --- async global->LDS copy (CDNA5 ASYNCcnt path), with sync fallback ------
#if defined(__AMDGCN__) && __has_builtin(__builtin_amdgcn_global_load_async_to_lds_b128)
#define USE_ASYNC_LDS 1
#else
#define USE_ASYNC_LDS 0
#endif

__device__ __forceinline__ void copy16_g2l(const float* __restrict__ g, float* l)
{
#if USE_ASYNC_LDS
    __builtin_amdgcn_global_load_async_to_lds_b128(
        (__attribute__((address_space(1))) v4i_gcc*)(void*)g,
        (__attribute__((address_space(3))) v4i_gcc*)(void*)l,
        /*imm offset=*/0, /*cpol=*/0);
#else
    *(float4*)l = *(const float4*)g;
#endif
}

__device__ __forceinline__ void wait_async_lds()
{
#if USE_ASYNC_LDS
#if __has_builtin(__builtin_amdgcn_s_wait_asynccnt)
    __builtin_amdgcn_s_wait_asynccnt(0);
#else
    asm volatile("s_wait_asynccnt 0x0" ::: "memory");
#endif
#endif
}

// C[M,N] = A[M,K] * W[N,K]^T   (A row-major MxK, W row-major NxK; both K-contiguous)
// Requires: M % TM == 0, N % TN == 0, K % TK == 0.
__global__ __launch_bounds__(128)
void wmma_gemm_f32(const float* __restrict__ A, const float* __restrict__ W,
                   const float* __restrict__ bias, const float* __restrict__ omega,
                   float* __restrict__ out0, float* __restrict__ out1,
                   int M, int N, int K, int mode)
{
    __shared__ float As[2][TM][LDS_LD];   // A tile, row-major, double buffered
    __shared__ float Bs[2][TN][LDS_LD];   // W tile, n-major: Bs[.][n][k] = W[nBlock+n][k0+k]

    const int tid  = threadIdx.x;
    const int wave = tid >> 5;               // 0..3 (wave32)
    const int lane = tid & 31;
    const int laneMod  = lane & 15;          // 0..15
    const int laneHalf = lane >> 4;          // 0 or 1

    const int mBlock = blockIdx.y * TM;
    const int nBlock = blockIdx.x * TN;
    const int waveM  = wave * 32;            // this wave's 32-row strip inside block tile

    v8f acc[2][4];
    #pragma unroll
    for (int mt = 0; mt < 2; ++mt)
        #pragma unroll
        for (int nt = 0; nt < 4; ++nt)
            #pragma unroll
            for (int r = 0; r < 8; ++r) acc[mt][nt][r] = 0.0f;

    // ---- tile stage: issues TM*TK/4/128 = 8 A chunks + 4 W chunks of b128 ----
    auto stage_tile = [&](int k0, int b) {
        #pragma unroll
        for (int i = 0; i < 8; ++i) {
            int q   = tid + i * 128;          // 0..1023
            int row = q >> 3;                 // 8 float4 per 32-float row
            int c4  = (q & 7) << 2;
            copy16_g2l(A + (size_t)(mBlock + row) * K + k0 + c4, &As[b][row][c4]);
        }
        #pragma unroll
        for (int i = 0; i < 4; ++i) {
            int q   = tid + i * 128;          // 0..511
            int row = q >> 3;
            int c4  = (q & 7) << 2;
            copy16_g2l(W + (size_t)(nBlock + row) * K + k0 + c4, &Bs[b][row][c4]);
        }
    };

    int buf = 0;
    stage_tile(0, buf);
    wait_async_lds();
    __syncthreads();

    for (int k0 = 0; k0 < K; k0 += TK) {
        const int nbuf = buf ^ 1;
        if (k0 + TK < K) stage_tile(k0 + TK, nbuf);   // prefetch next tile (async)

        // ---- 8 K-steps; per step: 2 A frags x 4 B frags = 8 WMMAs ----
        #pragma unroll
        for (int kk = 0; kk < TK; kk += 4) {
            // A frag (16x4): lanes 0-15: M=lane, K=kk+{0,1}; lanes 16-31: K=kk+{2,3}
            v2f a0, a1;
            a0.x = As[buf][waveM + laneMod][kk + 2 * laneHalf + 0];
            a0.y = As[buf][waveM + laneMod][kk + 2 * laneHalf + 1];
            a1.x = As[buf][waveM + 16 + laneMod][kk + 2 * laneHalf + 0];
            a1.y = As[buf][waveM + 16 + laneMod][kk + 2 * laneHalf + 1];
            #pragma unroll
            for (int nt = 0; nt < 4; ++nt) {
                // B frag (4x16): lane holds K=kk+2*laneHalf+{0,1}, N=nt*16+laneMod
                v2f bf;
                bf.x = Bs[buf][nt * 16 + laneMod][kk + 2 * laneHalf + 0];
                bf.y = Bs[buf][nt * 16 + laneMod][kk + 2 * laneHalf + 1];
                acc[0][nt] = __builtin_amdgcn_wmma_f32_16x16x4_f32(
                    false, a0, false, bf, (short)0, acc[0][nt], false, false);
                acc[1][nt] = __builtin_amdgcn_wmma_f32_16x16x4_f32(
                    false, a1, false, bf, (short)0, acc[1][nt], false, false);
            }
        }

        wait_async_lds();       // next tile landed in LDS
        __syncthreads();        // all waves done reading `buf`, async writes visible
        buf = nbuf;
    }

    // ---- epilogue: C/D layout: VGPR r -> row = r + 8*laneHalf, col = laneMod ----
    #pragma unroll
    for (int mt = 0; mt < 2; ++mt) {
        #pragma unroll
        for (int nt = 0; nt < 4; ++nt) {
            const int col = nBlock + nt * 16 + laneMod;
            #pragma unroll
            for (int r = 0; r < 8; ++r) {
                const int m = mBlock + waveM + mt * 16 + r + 8 * laneHalf;
                float val = acc[mt][nt][r];
                if (mode == EPI_PLAIN) {
                    if (bias) val += bias[col];
                    out0[(size_t)m * N + col] = val;
                } else if (mode == EPI_LAMBDA) {
                    val += bias[col];
                    float sig = 1.0f / (1.0f + expf(-val));
                    float sp  = log1pf(expf(omega[col]));          // softplus
                    out0[(size_t)m * N + col] = expf(-8.0f * sp * sig);
                } else { // EPI_SPLIT
                    if (col < D_INNER) out0[(size_t)m * D_INNER + col] = val;
                    else               out1[(size_t)m * D_INNER + (col - D_INNER)] = val;
                }
            }
        }
    }
}

// ---------------------------------------------------------------------------
// Sequential Householder scan. One block per batch, 1024 threads, 2 ch/thread.
// lam_h holds lam on entry; h is written in place (lam[t] read before h[t]
// written by the same thread; lam[.,0,.] is never consumed by the reference).
// ---------------------------------------------------------------------------
__global__ __launch_bounds__(1024)
void scan_kernel(const float* __restrict__ xi, const float* __restrict__ v,
                 float* __restrict__ lam_h)
{
    const int b   = blockIdx.x;
    const int tid = threadIdx.x;
    const int c0  = tid;
    const int c1  = tid + 1024;
    const size_t base = (size_t)b * SEQ * D_INNER;

    __shared__ float red[64];     // 32 waves x {vh, vv}
    __shared__ float bcast[2];

    float h0 = xi[base + c0];
    float h1 = xi[base + c1];
    lam_h[base + c0] = h0;
    lam_h[base + c1] = h1;

    const int wave = tid >> 5;
    const int lane = tid & 31;

    for (int t = 1; t < SEQ; ++t) {
        const size_t off = base + (size_t)t * D_INNER;
        const float v0 = v[off + c0],     v1 = v[off + c1];
        const float l0 = lam_h[off + c0], l1 = lam_h[off + c1];
        const float x0 = xi[off + c0],    x1 = xi[off + c1];

        float vh = v0 * h0 + v1 * h1;
        float vv = v0 * v0 + v1 * v1;
        #pragma unroll
        for (int s = 16; s > 0; s >>= 1) {   // wave32 butterfly
            vh += __shfl_xor(vh, s, 32);
            vv += __shfl_xor(vv, s, 32);
        }
        if (lane == 0) { red[wave] = vh; red[wave + 32] = vv; }
        __syncthreads();
        if (tid < 32) {
            float a  = red[tid];
            float bb = red[tid + 32];
            #pragma unroll
            for (int s = 16; s > 0; s >>= 1) {
                a  += __shfl_xor(a,  s, 32);
                bb += __shfl_xor(bb, s, 32);
            }
            if (tid == 0) { bcast[0] = a; bcast[1] = bb; }
        }
        __syncthreads();

        const float scale = 2.0f * bcast[0] / bcast[1];
        h0 = l0 * (h0 - v0 * scale) + (1.0f - l0) * x0;
        h1 = l1 * (h1 - v1 * scale) + (1.0f - l1) * x1;
        lam_h[off + c0] = h0;
        lam_h[off + c1] = h1;
        __syncthreads();   // red[] safe to overwrite next iteration
    }
}

// g = silu(z) * h, written over z
__global__ void silu_mul_kernel(const float* __restrict__ h, float* __restrict__ z_io,
                                size_t n)
{
    size_t i      = (size_t)blockIdx.x * blockDim.x + threadIdx.x;
    size_t stride = (size_t)gridDim.x * blockDim.x;
    for (; i < n; i += stride) {
        float zv = z_io[i];
        float s  = zv / (1.0f + expf(-zv));
        z_io[i] = s * h[i];
    }
}

// ---------------------------------------------------------------------------
extern "C" void kernel_launch(void* const* d_in, const int* in_sizes, int n_in,
                              void* d_out, int out_size, void* d_ws, size_t ws_size,
                              hipStream_t stream)
{
    const float* x     = (const float*)d_in[0];
    const float* omega = (const float*)d_in[1];   // 2048
    const float* Win   = (const float*)d_in[2];   // 4096 x 1024
    const float* Wl    = (const float*)d_in[3];   // 2048 x 2048
    const float* bl    = (const float*)d_in[4];
    const float* Wv    = (const float*)d_in[5];   // 2048 x 2048
    const float* bv    = (const float*)d_in[6];
    const float* Wout  = (const float*)d_in[7];   // 1024 x 2048
    float* out = (float*)d_out;

    const size_t BUF = (size_t)M_ROWS * D_INNER;  // 16,777,216 floats (64 MB)
    float* ws    = (float*)d_ws;
    float* xi    = ws;             // [M_ROWS, D_INNER]
    float* z     = ws + BUF;       // [M_ROWS, D_INNER] (later holds silu(z)*h)
    float* lam_h = ws + 2 * BUF;   // lam in, h out (in place)
    float* vbuf  = ws + 3 * BUF;   // total 256 MB of d_ws used

    dim3 blk(128);

    // 1) xz = x @ Win^T, split into xi | z
    {
        dim3 grd(4096 / TN, M_ROWS / TM);
        wmma_gemm_f32<<<grd, blk, 0, stream>>>(x, Win, nullptr, nullptr,
                                               xi, z, M_ROWS, 4096, D_MODEL, EPI_SPLIT);
    }
    // 2) lam = exp(-8*softplus(omega)*sigmoid(xi@Wl^T + bl))
    {
        dim3 grd(D_INNER / TN, M_ROWS / TM);
        wmma_gemm_f32<<<grd, blk, 0, stream>>>(xi, Wl, bl, omega,
                                               lam_h, nullptr, M_ROWS, D_INNER, D_INNER, EPI_LAMBDA);
    }
    // 3) v = xi @ Wv^T + bv
    {
        dim3 grd(D_INNER / TN, M_ROWS / TM);
        wmma_gemm_f32<<<grd, blk, 0, stream>>>(xi, Wv, bv, nullptr,
                                               vbuf, nullptr, M_ROWS, D_INNER, D_INNER, EPI_PLAIN);
    }
    // 4) sequential Householder scan (h overwrites lam buffer)
    scan_kernel<<<BATCH, 1024, 0, stream>>>(xi, vbuf, lam_h);

    // 5) g = silu(z) * h   (overwrites z)
    silu_mul_kernel<<<4096, 256, 0, stream>>>(lam_h, z, BUF);

    // 6) out = g @ Wout^T
    {
        dim3 grd(D_MODEL / TN, M_ROWS / TM);
        wmma_gemm_f32<<<grd, blk, 0, stream>>>(z, Wout, nullptr, nullptr,
                                               out, nullptr, M_ROWS, D_MODEL, D_INNER, EPI_PLAIN);
    }
}